// SpanEmbedder_44598940402063
// MI455X (gfx1250) — compile-verified
//
#include <hip/hip_runtime.h>

// Problem constants (from reference)
#define S_DIM 64
#define N_DIM 128
#define L_DIM 10
#define D_DIM 300
#define H_DIM 150
#define OUTC  (3 * D_DIM)   // 900

typedef __attribute__((ext_vector_type(2))) float v2f;
typedef __attribute__((ext_vector_type(8))) float v8f;

// ---------------------------------------------------------------------------
// Kernel 1: scores[m] = relu(emb_table[tokens[m]] @ W1 + b1) @ W2 + b2
// Grid: (S*N)/16 = 512 blocks. Block: 320 threads = 10 waves.
// Wave `wv` computes the 16x16 C tile (M = 16 tokens, N = H-columns
// wv*16..wv*16+15) via v_wmma_f32_16x16x4_f32 over K = D in steps of 4.
// ---------------------------------------------------------------------------
__global__ __launch_bounds__(320)
void score_kernel(const int* __restrict__ tokens,
                  const float* __restrict__ emb_table,
                  const float* __restrict__ W1,
                  const float* __restrict__ b1,
                  const float* __restrict__ W2,
                  const float* __restrict__ b2,
                  float* __restrict__ scores)
{
    __shared__ float smem[10][16];

    const int lane = threadIdx.x & 31;
    const int wv   = threadIdx.x >> 5;      // 0..9  -> N tile
    const int half = lane >> 4;             // 0/1   (lanes 0-15 vs 16-31)
    const int l16  = lane & 15;

    // A rows: M = mtile*16 + l16 (token index)
    const int m   = blockIdx.x * 16 + l16;
    const int tok = tokens[m];
    const float* arow = emb_table + (long)tok * D_DIM;

    // B cols: N = wv*16 + l16 (H column), clamp tail tile (H=150 < 160)
    const int nn    = wv * 16 + l16;
    const int n_eff = nn < (H_DIM - 1) ? nn : (H_DIM - 1);
    const float w2v = (nn < H_DIM) ? W2[nn] : 0.0f;
    const float b1v = (nn < H_DIM) ? b1[nn] : 0.0f;

    v8f acc = {};
    // K loop over D=300 in steps of 4: 75 WMMAs per wave.
    // ISA A layout (32-bit, 16x4): lanes 0-15 hold K={0,1}, lanes 16-31 K={2,3}
    // ISA B layout (4x16): VGPR0 = row K (0 / 2), VGPR1 = row K+1 (1 / 3)
    for (int k = 0; k < D_DIM; k += 4) {
        const int ka = k + 2 * half;
        float2 av = *reinterpret_cast<const float2*>(arow + ka); // 8B aligned
        v2f a; a[0] = av.x; a[1] = av.y;
        v2f b;
        b[0] = W1[(long)ka * H_DIM + n_eff];
        b[1] = W1[(long)(ka + 1) * H_DIM + n_eff];
        acc = __builtin_amdgcn_wmma_f32_16x16x4_f32(
            /*neg_a=*/false, a, /*neg_b=*/false, b,
            /*c_mod=*/(short)0, acc, /*reuse_a=*/false, /*reuse_b=*/false);
    }

    // C layout: VGPR v, lanes 0-15 -> M=v, lanes 16-31 -> M=v+8; lane = N.
    // q[v] = relu(c + b1[n]) * W2[n]; reduce across the 16 lanes of each half.
    float q[8];
#pragma unroll
    for (int v = 0; v < 8; ++v) {
        float h = acc[v] + b1v;
        h = h > 0.0f ? h : 0.0f;
        q[v] = h * w2v;
    }
#pragma unroll
    for (int off = 1; off < 16; off <<= 1) {
#pragma unroll
        for (int v = 0; v < 8; ++v)
            q[v] += __shfl_xor(q[v], off, 32);
    }
    // lanes 0 and 16 now hold the 8 sums for M = v (+8*half)
    if (l16 == 0) {
#pragma unroll
        for (int v = 0; v < 8; ++v)
            smem[wv][v + 8 * half] = q[v];
    }
    __syncthreads();

    // Reduce across the 10 N-tiles, add b2, store
    if (threadIdx.x < 16) {
        float s = 0.0f;
#pragma unroll
        for (int t = 0; t < 10; ++t) s += smem[t][threadIdx.x];
        scores[blockIdx.x * 16 + threadIdx.x] = s + b2[0];
    }
}

// ---------------------------------------------------------------------------
// Kernel 2: per (s,n) block, wave w emits output row w:
//   out[s,n,w,:] = [ emb[s,n,:], emb[s,n+w,:], softmax_{k<=w}(sc) @ emb_win ]
// masked to zero when n+w >= N. Output-write bound (295 MB).
// Grid: S*N = 8192 blocks. Block: 320 threads = 10 waves (one per w).
// ---------------------------------------------------------------------------
__global__ __launch_bounds__(320)
void span_kernel(const int* __restrict__ tokens,
                 const float* __restrict__ emb_table,
                 const float* __restrict__ scores,
                 float* __restrict__ out)
{
    const int bid  = blockIdx.x;          // (s,n) flat
    const int s    = bid / N_DIM;
    const int n    = bid % N_DIM;
    const int lane = threadIdx.x & 31;
    const int w    = threadIdx.x >> 5;    // 0..9, wave-uniform

    float* orow = out + ((long)bid * L_DIM + w) * OUTC;

    const bool rowvalid = (n + w) < N_DIM;   // wave-uniform branch
    if (!rowvalid) {
        for (int d = lane; d < OUTC; d += 32) orow[d] = 0.0f;
        return;
    }

    // Masked softmax over k = 0..w (all those positions are in-range here).
    float p[L_DIM];
    float mx = -3.402823466e+38f;
#pragma unroll
    for (int k = 0; k < L_DIM; ++k) {
        if (k <= w) {
            p[k] = scores[s * N_DIM + n + k];
            mx = fmaxf(mx, p[k]);
        } else {
            p[k] = 0.0f;
        }
    }
    float denom = 0.0f;
#pragma unroll
    for (int k = 0; k < L_DIM; ++k) {
        if (k <= w) { p[k] = __expf(p[k] - mx); denom += p[k]; }
    }
    const float inv = 1.0f / denom;
#pragma unroll
    for (int k = 0; k < L_DIM; ++k) p[k] *= inv;

    // Hoist the gathered row pointers (L2-resident emb_table rows).
    const float* rows[L_DIM];
#pragma unroll
    for (int k = 0; k < L_DIM; ++k) {
        const int kk = (k <= w) ? k : 0;
        rows[k] = emb_table + (long)tokens[s * N_DIM + n + kk] * D_DIM;
    }
    const float* rfirst = rows[0];               // k=0 -> emb[s,n,:]
    const float* rlast  = rows[w];               // emb[s,n+w,:]

    for (int d = lane; d < D_DIM; d += 32) {
        float hv = 0.0f;
#pragma unroll
        for (int k = 0; k < L_DIM; ++k) {
            if (k <= w) hv = fmaf(p[k], rows[k][d], hv);
        }
        orow[d]             = rfirst[d];
        orow[D_DIM + d]     = rlast[d];
        orow[2 * D_DIM + d] = hv;
    }
}

extern "C" void kernel_launch(void* const* d_in, const int* in_sizes, int n_in,
                              void* d_out, int out_size, void* d_ws, size_t ws_size,
                              hipStream_t stream) {
    const int*   tokens    = (const int*)  d_in[0];
    const float* emb_table = (const float*)d_in[1];
    const float* W1        = (const float*)d_in[2];
    const float* b1        = (const float*)d_in[3];
    const float* W2        = (const float*)d_in[4];
    const float* b2        = (const float*)d_in[5];
    float* out    = (float*)d_out;
    float* scores = (float*)d_ws;   // S*N = 8192 floats = 32 KB scratch

    const int mtiles = (S_DIM * N_DIM) / 16;          // 512
    score_kernel<<<mtiles, 320, 0, stream>>>(tokens, emb_table, W1, b1, W2, b2, scores);
    span_kernel<<<S_DIM * N_DIM, 320, 0, stream>>>(tokens, emb_table, scores, out);
}